// GCN_88287347737168
// MI455X (gfx1250) — compile-verified
//
#include <hip/hip_runtime.h>

typedef __attribute__((ext_vector_type(2))) float v2f;
typedef __attribute__((ext_vector_type(8))) float v8f;

#define BN_EPS 1e-5f

// ---------------------------------------------------------------------------
// GEMM  H[N,CO] = A[N,128] * W[128,CO]   via V_WMMA_F32_16X16X4_F32
// One wave computes a 16x32 strip (two 16x16 C tiles sharing the A fragment).
// A frag (16x4 f32): lane%16 = M row; lane/16 selects K pair {k0,k0+1} vs
// {k0+2,k0+3}; VGPR0/1 = the two K's of the pair.
// B frag (4x16 f32): lane%16 = N col; lane/16 selects K pair (mirrored).
// C/D (16x16 f32): VGPR r -> M = r + 8*(lane/16); N = lane%16.
// ---------------------------------------------------------------------------
template <int CO>
__global__ void gemm_wmma_f32(const float* __restrict__ A,
                              const float* __restrict__ W,
                              float* __restrict__ H, int N) {
  constexpr int K = 128;
  constexpr int pairsN = CO / 32;  // 16x32 strips along the output columns
  const int lane = threadIdx.x & 31;
  const int wave = threadIdx.x >> 5;
  const int l16  = lane & 15;
  const int hi   = lane >> 4;  // 0: lanes 0-15, 1: lanes 16-31

  const int tilesM = (N + 15) >> 4;
  const int tile = blockIdx.x * (blockDim.x >> 5) + wave;
  if (tile >= tilesM * pairsN) return;
  const int tm = tile / pairsN;
  const int tp = tile % pairsN;

  int row = tm * 16 + l16;
  int rowc = row < N ? row : N - 1;  // clamp (N%16==0 in practice)
  const float* __restrict__ Arow = A + (size_t)rowc * K;
  const float* __restrict__ Wc   = W + tp * 32 + l16;

  v8f c0 = {};
  v8f c1 = {};
#pragma unroll
  for (int k0 = 0; k0 < K; k0 += 4) {
    const int ka = k0 + hi * 2;
    v2f a;
    a[0] = Arow[ka];
    a[1] = Arow[ka + 1];
    v2f b0, b1;
    b0[0] = Wc[(size_t)ka * CO];
    b0[1] = Wc[(size_t)(ka + 1) * CO];
    b1[0] = Wc[(size_t)ka * CO + 16];
    b1[1] = Wc[(size_t)(ka + 1) * CO + 16];
    c0 = __builtin_amdgcn_wmma_f32_16x16x4_f32(false, a, false, b0, (short)0,
                                               c0, false, false);
    c1 = __builtin_amdgcn_wmma_f32_16x16x4_f32(false, a, false, b1, (short)0,
                                               c1, false, false);
  }

  const int baseRow = tm * 16 + hi * 8;
  float* __restrict__ Hp = H + (size_t)baseRow * CO + tp * 32 + l16;
#pragma unroll
  for (int r = 0; r < 8; ++r) {
    if (baseRow + r < N) {
      Hp[(size_t)r * CO]      = c0[r];
      Hp[(size_t)r * CO + 16] = c1[r];
    }
  }
}

// ---------------------------------------------------------------------------
// Edge scatter: out[dst[e]] += H[src[e]] * w[e].
// Each thread owns one float4 channel-group of one edge; lanes of a wave share
// the same edge (C=128) or split into 2 edges (C=64), so index loads coalesce
// and the 51MB feature matrix is served from L2. Atomics -> global_atomic_add_f32.
// ---------------------------------------------------------------------------
__global__ void edge_scatter(const float* __restrict__ H,
                             const int* __restrict__ src,
                             const int* __restrict__ dst,
                             const float* __restrict__ ew,
                             float* __restrict__ out, int E, int C,
                             int gshift) {
  const long long t = (long long)blockIdx.x * blockDim.x + threadIdx.x;
  const long long total = (long long)E << gshift;
  if (t >= total) return;
  const int e = (int)(t >> gshift);
  const int g = (int)t & ((1 << gshift) - 1);
  const int s = src[e];
  const int d = dst[e];
  const float w = ew[e];
  const float4 v = *reinterpret_cast<const float4*>(H + (size_t)s * C + g * 4);
  float* __restrict__ o = out + (size_t)d * C + g * 4;
  atomicAdd(o + 0, v.x * w);
  atomicAdd(o + 1, v.y * w);
  atomicAdd(o + 2, v.z * w);
  atomicAdd(o + 3, v.w * w);
}

// ---------------------------------------------------------------------------
// Per-channel sum / sum-of-squares (blockDim.x == C, rows strided over blocks)
// ---------------------------------------------------------------------------
__global__ void channel_stats(const float* __restrict__ X,
                              float* __restrict__ st, int N, int C) {
  const int c = threadIdx.x;
  float s = 0.f, s2 = 0.f;
  for (int r = blockIdx.x; r < N; r += gridDim.x) {
    const float v = X[(size_t)r * C + c];
    s += v;
    s2 += v * v;
  }
  atomicAdd(&st[c], s);
  atomicAdd(&st[C + c], s2);
}

// ---------------------------------------------------------------------------
// Fused BN (training-mode stats) + ReLU.  conv bias cancels inside BN.
// ---------------------------------------------------------------------------
__global__ void bn_relu(const float* __restrict__ X,
                        const float* __restrict__ st,
                        const float* __restrict__ g,
                        const float* __restrict__ be, float* __restrict__ Y,
                        long long total, int C, float invN) {
  const long long t = (long long)blockIdx.x * blockDim.x + threadIdx.x;
  if (t >= total) return;
  const int c = (int)t & (C - 1);
  const float mean = st[c] * invN;
  const float var = st[C + c] * invN - mean * mean;
  const float rs = rsqrtf(var + BN_EPS);
  float y = (X[t] - mean) * rs * g[c] + be[c];
  Y[t] = y > 0.f ? y : 0.f;
}

// out[n][c] = b[c]  (seed the final scatter target with the layer-3 bias)
__global__ void init_bias(float* __restrict__ out, const float* __restrict__ b,
                          long long total, int C) {
  const long long t = (long long)blockIdx.x * blockDim.x + threadIdx.x;
  if (t >= total) return;
  out[t] = b[(int)t & (C - 1)];
}

static inline int blocks_for(long long total, int bs) {
  return (int)((total + bs - 1) / bs);
}

extern "C" void kernel_launch(void* const* d_in, const int* in_sizes, int n_in,
                              void* d_out, int out_size, void* d_ws,
                              size_t ws_size, hipStream_t stream) {
  const float* x   = (const float*)d_in[0];
  const int*   ei  = (const int*)d_in[1];
  const float* ew  = (const float*)d_in[2];
  const float* W1  = (const float*)d_in[3];
  // d_in[4] = b1: cancels under batchnorm
  const float* g1  = (const float*)d_in[5];
  const float* be1 = (const float*)d_in[6];
  const float* W2  = (const float*)d_in[7];
  // d_in[8] = b2: cancels under batchnorm
  const float* g2  = (const float*)d_in[9];
  const float* be2 = (const float*)d_in[10];
  const float* W3  = (const float*)d_in[11];
  const float* b3  = (const float*)d_in[12];

  const int C = 128, CO = 64;
  const int N = in_sizes[0] / C;
  const int E = in_sizes[1] / 2;
  const int* src = ei;
  const int* dst = ei + E;

  float* bufA = (float*)d_ws;                 // N*128
  float* bufB = bufA + (size_t)N * C;         // N*128
  float* st   = bufB + (size_t)N * C;         // 256 floats
  const float invN = 1.0f / (float)N;

  const int BS = 256;
  const int tilesM = (N + 15) / 16;
  const int waves128 = tilesM * (C / 32);
  const int waves64  = tilesM * (CO / 32);
  const long long nc  = (long long)N * C;
  const long long nco = (long long)N * CO;
  const long long es128 = (long long)E << 5;  // E * (128/4)
  const long long es64  = (long long)E << 4;  // E * (64/4)

  // ---- Layer 1: h = relu(BN(scatter(x@W1))) -------------------------------
  gemm_wmma_f32<128><<<blocks_for(waves128, 8), BS, 0, stream>>>(x, W1, bufA, N);
  hipMemsetAsync(bufB, 0, nc * sizeof(float), stream);
  edge_scatter<<<blocks_for(es128, BS), BS, 0, stream>>>(bufA, src, dst, ew,
                                                         bufB, E, C, 5);
  hipMemsetAsync(st, 0, 2 * C * sizeof(float), stream);
  channel_stats<<<1024, C, 0, stream>>>(bufB, st, N, C);
  bn_relu<<<blocks_for(nc, BS), BS, 0, stream>>>(bufB, st, g1, be1, bufA, nc, C,
                                                 invN);

  // ---- Layer 2 ------------------------------------------------------------
  gemm_wmma_f32<128><<<blocks_for(waves128, 8), BS, 0, stream>>>(bufA, W2, bufB,
                                                                 N);
  hipMemsetAsync(bufA, 0, nc * sizeof(float), stream);
  edge_scatter<<<blocks_for(es128, BS), BS, 0, stream>>>(bufB, src, dst, ew,
                                                         bufA, E, C, 5);
  hipMemsetAsync(st, 0, 2 * C * sizeof(float), stream);
  channel_stats<<<1024, C, 0, stream>>>(bufA, st, N, C);
  bn_relu<<<blocks_for(nc, BS), BS, 0, stream>>>(bufA, st, g2, be2, bufB, nc, C,
                                                 invN);

  // ---- Layer 3: out = scatter(h@W3) + b3 ----------------------------------
  gemm_wmma_f32<64><<<blocks_for(waves64, 8), BS, 0, stream>>>(bufB, W3, bufA,
                                                               N);
  float* out = (float*)d_out;
  init_bias<<<blocks_for(nco, BS), BS, 0, stream>>>(out, b3, nco, CO);
  edge_scatter<<<blocks_for(es64, BS), BS, 0, stream>>>(bufA, src, dst, ew, out,
                                                        E, CO, 4);
}